// DeformerNetBimanual_44169443672120
// MI455X (gfx1250) — compile-verified
//
#include <hip/hip_runtime.h>
#include <hip/hip_bf16.h>

// ---------------------------------------------------------------------------
// DeformerNetBimanual forward for gfx1250 (MI455X).
// All GEMM-shaped work runs through a strided-batched WMMA kernel using
// V_WMMA_F32_16X16X4_F32 (fp32 in / fp32 acc -> matches the fp32 reference).
// ---------------------------------------------------------------------------

#define EPSV 1e-5f
#define CDIV(a, b) (((a) + (b) - 1) / (b))

static const int B_ = 64;
static const int N_ = 1024;

typedef __attribute__((ext_vector_type(2))) float v2f;
typedef __attribute__((ext_vector_type(8))) float v8f;

#if __has_builtin(__builtin_amdgcn_wmma_f32_16x16x4_f32)
#define WMMA4(a, b, c)                                                       \
  __builtin_amdgcn_wmma_f32_16x16x4_f32(false, (a), false, (b), (short)0, (c), \
                                        false, false)
#else
static __device__ inline v8f wmma_fb(v2f a, v2f b, v8f c) {
  c[0] += a.x * b.x + a.y * b.y;  // placeholder fallback (diagnosed via wmma==0)
  return c;
}
#define WMMA4(a, b, c) wmma_fb(a, b, c)
#endif

// ---------------------------------------------------------------------------
// Generic strided batched GEMM:  D[z][m][n] = sum_k A[z](m,k) * B[z](k,n) (+bias[n])
//   A element (m,k) at A + z*batchA + m*As_m + k*As_k
//   B element (k,n) at B + z*batchB + k*Bs_k + n*Bs_n
//   D row-major, leading dim ldD, at D + z*batchD
// K-strides are compile-time template params (ASK_/BSK_, 0 = use runtime arg)
// so the unrolled main loop uses immediate load offsets off one base pointer.
// M must be a multiple of 16 (true for every call site). Columns >= N are
// computed on clamped addresses and simply not stored (column n of D depends
// only on lane n's B fragment, so stored columns are exact).
// One wave computes one 16x16 tile, K stepped by 4, with a 2-deep software
// pipeline (quads q+1,q+2 in flight while quad q computes) + guarded tail.
// Prefetch once per 8 quads (one prefetch covers a full cacheline).
// Fragment layouts per CDNA5 ISA 7.12.2:
//   A: lane L (L&15 = row), half (L>>4) selects K pair {k0,k0+1} vs {k0+2,k0+3}
//   B: lane L (L&15 = col), same K-pair split across lane halves
//   C/D: VGPR v, lanes 0-15 -> row v, lanes 16-31 -> row v+8
// ---------------------------------------------------------------------------
template <int ASK_, int BSK_>
__global__ __launch_bounds__(256) void k_gemm_wmma(
    const float* __restrict__ A, const float* __restrict__ Bm,
    const float* __restrict__ bias, float* __restrict__ D,
    long long M, int N, int K,
    long long As_m, long long As_k_rt, long long Bs_k_rt, long long Bs_n, int ldD,
    long long batchA, long long batchB, long long batchD) {
  const long long ASK = ASK_ ? ASK_ : As_k_rt;  // folds to constant when ASK_>0
  const long long BSK = BSK_ ? BSK_ : Bs_k_rt;

  const int lane = threadIdx.x & 31;
  const int wave = threadIdx.x >> 5;
  const int tN = (N + 15) >> 4;
  const long long tM = M >> 4;
  const long long tile = (long long)blockIdx.x * (blockDim.x >> 5) + wave;
  if (tile >= tM * (long long)tN) return;  // wave-uniform: EXEC stays all-1s
  const long long tm = tile / tN;
  const int tn = (int)(tile - tm * tN);
  const long long m0 = tm << 4;
  const int n0 = tn << 4;

  A += (long long)blockIdx.y * batchA;
  Bm += (long long)blockIdx.y * batchB;
  D += (long long)blockIdx.y * batchD;

  const int kh = lane >> 4;  // K-pair half
  const int lm = lane & 15;
  const long long am = m0 + lm;             // A row handled by this lane (< M)
  const int bn = n0 + lm;                   // output column handled by this lane
  const int bnc = (bn < N) ? bn : (N - 1);  // clamped: loads always in-bounds

  const float* pa = A + am * As_m + (long long)(kh << 1) * ASK;
  const float* pb = Bm + (long long)bnc * Bs_n + (long long)(kh << 1) * BSK;

  auto lda = [&](int q) {
    v2f a;
    a.x = pa[4ll * ASK * q];
    a.y = pa[4ll * ASK * q + ASK];
    return a;
  };
  auto ldb = [&](int q) {
    v2f b;
    b.x = pb[4ll * BSK * q];
    b.y = pb[4ll * BSK * q + BSK];
    return b;
  };

  v8f acc = {};
  const int K4 = K & ~3;
  const int nq = K4 >> 2;  // number of full K-quads
  if (nq > 0) {
    v2f a0 = lda(0), b0 = ldb(0);
    v2f a1 = {}, b1 = {};
    if (nq > 1) { a1 = lda(1); b1 = ldb(1); }
    int q = 0;
    for (; q + 2 < nq; ++q) {
      v2f a2 = lda(q + 2), b2 = ldb(q + 2);
      if ((q & 7) == 0) {  // one cacheline-granular prefetch per 8 quads
        __builtin_prefetch(pa + 4ll * ASK * (q + 8), 0, 1);  // global_prefetch_b8
        __builtin_prefetch(pb + 4ll * BSK * (q + 8), 0, 1);
      }
      acc = WMMA4(a0, b0, acc);
      a0 = a1; b0 = b1;
      a1 = a2; b1 = b2;
    }
    acc = WMMA4(a0, b0, acc);
    if (nq > 1) acc = WMMA4(a1, b1, acc);
  }
  if (K & 3) {  // guarded tail quad (K = 3 / 6 / 67 / 131 call sites)
    const long long ba = 4ll * ASK * nq;
    const long long bb = 4ll * BSK * nq;
    const int ka = K4 + (kh << 1);
    const int kb = ka + 1;
    v2f a, b;
    a.x = (ka < K) ? pa[ba] : 0.0f;
    a.y = (kb < K) ? pa[ba + ASK] : 0.0f;
    b.x = (ka < K) ? pb[bb] : 0.0f;
    b.y = (kb < K) ? pb[bb + BSK] : 0.0f;
    acc = WMMA4(a, b, acc);
  }
#pragma unroll
  for (int v = 0; v < 8; ++v) {
    const long long m = m0 + v + (kh << 3);
    if (bn < N) {
      float x = acc[v];
      if (bias) x += bias[bn];
      D[m * (long long)ldD + bn] = x;
    }
  }
}

// --- layout helpers --------------------------------------------------------
// in [B, Cin, Np] -> out [B, Np, Csel] (first Csel channels)
__global__ void k_chw_to_hwc_sel(const float* __restrict__ in, float* __restrict__ out,
                                 int Cin, int Csel, int Np) {
  long long idx = (long long)blockIdx.x * blockDim.x + threadIdx.x;
  long long total = (long long)B_ * Np * Csel;
  if (idx >= total) return;
  int c = (int)(idx % Csel);
  long long t = idx / Csel;
  int n = (int)(t % Np);
  int b = (int)(t / Np);
  out[idx] = in[((long long)b * Cin + c) * Np + n];
}

// inverse kernel density per point: invden[b,n] = 1 / mean_m exp(-d2/(2bw^2))/(2.5bw)
__global__ void k_density(const float* __restrict__ P, float* __restrict__ invden,
                          int Np, float bw) {
  int idx = blockIdx.x * blockDim.x + threadIdx.x;
  if (idx >= B_ * Np) return;
  int b = idx / Np, n = idx % Np;
  const float* pb = P + (long long)b * Np * 3;
  float x = pb[n * 3], y = pb[n * 3 + 1], z = pb[n * 3 + 2];
  float inv2 = 1.0f / (2.0f * bw * bw);
  float s = 0.0f;
  for (int m = 0; m < Np; ++m) {
    float dx = pb[m * 3] - x, dy = pb[m * 3 + 1] - y, dz = pb[m * 3 + 2] - z;
    s += __expf(-(dx * dx + dy * dy + dz * dz) * inv2);
  }
  float density = (s / (2.5f * bw)) / (float)Np;
  invden[idx] = 1.0f / density;
}

// farthest point sampling: one block per batch, blockDim == Np (power of 2)
__global__ void k_fps(const float* __restrict__ P, int* __restrict__ outidx,
                      int Np, int npoint) {
  __shared__ float s_red[1024];
  __shared__ int s_idx[1024];
  __shared__ float s_cent[3];
  int b = blockIdx.x, t = threadIdx.x;
  const float* pb = P + (long long)b * Np * 3;
  float px = pb[t * 3], py = pb[t * 3 + 1], pz = pb[t * 3 + 2];
  float dist = 1e10f;
  int far = 0;
  for (int s = 0; s < npoint; ++s) {
    if (t == 0) outidx[b * npoint + s] = far;
    if (t == far) { s_cent[0] = px; s_cent[1] = py; s_cent[2] = pz; }
    __syncthreads();
    float dx = px - s_cent[0], dy = py - s_cent[1], dz = pz - s_cent[2];
    dist = fminf(dist, dx * dx + dy * dy + dz * dz);
    s_red[t] = dist; s_idx[t] = t;
    __syncthreads();
    for (int o = Np >> 1; o > 0; o >>= 1) {
      if (t < o && s_red[t + o] > s_red[t]) { s_red[t] = s_red[t + o]; s_idx[t] = s_idx[t + o]; }
      __syncthreads();
    }
    far = s_idx[0];
    __syncthreads();
  }
}

__global__ void k_gather3(const float* __restrict__ P, const int* __restrict__ idx,
                          float* __restrict__ out, int Np, int S) {
  int i = blockIdx.x * blockDim.x + threadIdx.x;
  if (i >= B_ * S) return;
  int b = i / S;
  const float* p = P + ((long long)b * Np + idx[i]) * 3;
  out[i * 3 + 0] = p[0]; out[i * 3 + 1] = p[1]; out[i * 3 + 2] = p[2];
}

// k nearest neighbors by repeated min extraction; one block of 256 per (b,s)
__global__ void k_knn(const float* __restrict__ P, const float* __restrict__ cen,
                      int* __restrict__ knni, int Np, int S, int ns) {
  extern __shared__ float sdyn[];  // Np distances
  __shared__ float s_min[256];
  __shared__ int s_mi[256];
  long long bs = blockIdx.x;
  int b = (int)(bs / S);
  int t = threadIdx.x;
  const float* pb = P + (long long)b * Np * 3;
  float cx = cen[bs * 3], cy = cen[bs * 3 + 1], cz = cen[bs * 3 + 2];
  for (int n = t; n < Np; n += 256) {
    float dx = pb[n * 3] - cx, dy = pb[n * 3 + 1] - cy, dz = pb[n * 3 + 2] - cz;
    sdyn[n] = dx * dx + dy * dy + dz * dz;
  }
  __syncthreads();
  for (int j = 0; j < ns; ++j) {
    float best = 1e30f; int bi = 0x7fffffff;
    for (int n = t; n < Np; n += 256)
      if (sdyn[n] < best) { best = sdyn[n]; bi = n; }
    s_min[t] = best; s_mi[t] = bi;
    __syncthreads();
    for (int o = 128; o > 0; o >>= 1) {
      if (t < o) {
        float ov = s_min[t + o]; int oi = s_mi[t + o];
        if (ov < s_min[t] || (ov == s_min[t] && oi < s_mi[t])) { s_min[t] = ov; s_mi[t] = oi; }
      }
      __syncthreads();
    }
    if (t == 0) { knni[bs * ns + j] = s_mi[0]; sdyn[s_mi[0]] = 1e30f; }
    __syncthreads();
  }
}

// grouping: feats=[gxyz | gathered points], gxyz centered, gden gathered
__global__ void k_group(const float* __restrict__ P, const float* __restrict__ F,
                        const float* __restrict__ cen, const float* __restrict__ invden,
                        const int* __restrict__ knni, float* __restrict__ feats,
                        float* __restrict__ gxyz, float* __restrict__ gden,
                        int Np, int S, int ns, int Cpts, int Cf) {
  long long idx = (long long)blockIdx.x * blockDim.x + threadIdx.x;
  long long total = (long long)B_ * S * ns;
  if (idx >= total) return;
  long long bs = idx / ns;
  int b = (int)(bs / S);
  int n = knni[idx];
  const float* pp = P + ((long long)b * Np + n) * 3;
  const float* cc = cen + bs * 3;
  float gx = pp[0] - cc[0], gy = pp[1] - cc[1], gz = pp[2] - cc[2];
  gxyz[idx * 3 + 0] = gx; gxyz[idx * 3 + 1] = gy; gxyz[idx * 3 + 2] = gz;
  float* fo = feats + idx * Cf;
  fo[0] = gx; fo[1] = gy; fo[2] = gz;
  const float* fp = F + ((long long)b * Np + n) * Cpts;
  for (int c = 0; c < Cpts; ++c) fo[3 + c] = fp[c];
  gden[idx] = invden[(long long)b * Np + n];
}

// group_all: feats=[P | F] (uncentered), gxyz=P, gden=invden
__global__ void k_group_all(const float* __restrict__ P, const float* __restrict__ F,
                            const float* __restrict__ invden, float* __restrict__ feats,
                            float* __restrict__ gxyz, float* __restrict__ gden,
                            int Np, int Cpts, int Cf) {
  long long idx = (long long)blockIdx.x * blockDim.x + threadIdx.x;
  long long total = (long long)B_ * Np;
  if (idx >= total) return;
  int b = (int)(idx / Np);
  int n = (int)(idx % Np);
  const float* pp = P + ((long long)b * Np + n) * 3;
  gxyz[idx * 3 + 0] = pp[0]; gxyz[idx * 3 + 1] = pp[1]; gxyz[idx * 3 + 2] = pp[2];
  float* fo = feats + idx * Cf;
  fo[0] = pp[0]; fo[1] = pp[1]; fo[2] = pp[2];
  const float* fp = F + ((long long)b * Np + n) * Cpts;
  for (int c = 0; c < Cpts; ++c) fo[3 + c] = fp[c];
  gden[idx] = invden[(long long)b * Np + n];
}

// --- batch norm (stats over rows, per column) ------------------------------
__global__ void k_zero(float* __restrict__ p, int n) {
  for (int i = threadIdx.x; i < n; i += 256) p[i] = 0.0f;
}
__global__ void k_colstats(const float* __restrict__ X, float* __restrict__ st,
                           long long M, int C) {
  int idx = blockIdx.x * blockDim.x + threadIdx.x;
  if (idx >= C * 256) return;
  int c = idx % C;
  int g = idx / C;
  float s = 0.0f, q = 0.0f;
  for (long long r = g; r < M; r += 256) {
    float v = X[r * C + c];
    s += v; q += v * v;
  }
  atomicAdd(&st[c], s);
  atomicAdd(&st[C + c], q);
}
__global__ void k_bn_apply_relu(float* __restrict__ X, const float* __restrict__ st,
                                const float* __restrict__ g, const float* __restrict__ be,
                                long long M, int C) {
  long long idx = (long long)blockIdx.x * blockDim.x + threadIdx.x;
  if (idx >= M * C) return;
  int c = (int)(idx % C);
  float mean = st[c] / (float)M;
  float var = st[C + c] / (float)M - mean * mean;
  float y = (X[idx] - mean) * rsqrtf(var + EPSV) * g[c] + be[c];
  X[idx] = fmaxf(y, 0.0f);
}

// h[b,s,k,c] *= gden[b,s,k] / max_k gden[b,s,:]   (one block per (b,s))
__global__ void k_den_scale(float* __restrict__ h, const float* __restrict__ gden,
                            int ns, int C) {
  __shared__ float smax[256];
  long long bs = blockIdx.x;
  const float* gd = gden + bs * ns;
  int t = threadIdx.x;
  float m = -1e30f;
  for (int k = t; k < ns; k += 256) m = fmaxf(m, gd[k]);
  smax[t] = m;
  __syncthreads();
  for (int o = 128; o > 0; o >>= 1) {
    if (t < o) smax[t] = fmaxf(smax[t], smax[t + o]);
    __syncthreads();
  }
  float inv = 1.0f / smax[0];
  float* hb = h + bs * (long long)ns * C;
  for (long long i = t; i < (long long)ns * C; i += 256) {
    int k = (int)(i / C);
    hb[i] *= gd[k] * inv;
  }
}

// group norm over channel axis (one block per sample row) + ReLU
__global__ void k_gn_relu(float* __restrict__ X, const float* __restrict__ g,
                          const float* __restrict__ be, int C) {
  __shared__ float ssum[256], ssq[256];
  int b = blockIdx.x, t = threadIdx.x;
  float* x = X + (long long)b * C;
  float s = 0.0f, q = 0.0f;
  for (int c = t; c < C; c += 256) { float v = x[c]; s += v; q += v * v; }
  ssum[t] = s; ssq[t] = q;
  __syncthreads();
  for (int o = 128; o > 0; o >>= 1) {
    if (t < o) { ssum[t] += ssum[t + o]; ssq[t] += ssq[t + o]; }
    __syncthreads();
  }
  float mean = ssum[0] / C;
  float var = ssq[0] / C - mean * mean;
  float r = rsqrtf(var + EPSV);
  for (int c = t; c < C; c += 256) x[c] = fmaxf((x[c] - mean) * r * g[c] + be[c], 0.0f);
}

__global__ void k_concat(const float* __restrict__ x, const float* __restrict__ gf,
                         float* __restrict__ out) {
  int idx = blockIdx.x * blockDim.x + threadIdx.x;
  if (idx >= B_ * 512) return;
  int c = idx % 512, b = idx / 512;
  out[idx] = (c < 256) ? x[b * 256 + c] : gf[b * 256 + (c - 256)];
}

__global__ void k_fc5(const float* __restrict__ X, const float* __restrict__ W,
                      const float* __restrict__ bb, float* __restrict__ out) {
  int idx = blockIdx.x * blockDim.x + threadIdx.x;
  if (idx >= B_ * 6) return;
  int j = idx % 6, b = idx / 6;
  float s = bb[j];
  for (int c = 0; c < 64; ++c) s += X[b * 64 + c] * W[j * 64 + c];
  out[idx] = s;
}

// ---------------------------------------------------------------------------
// Host orchestration.
// Input flattening (setup_inputs dict order, then params pytree in sorted-key
// DFS): 0 xyz, 1 xyz_goal, 2..9 fc{1,3,4,5}{W,b}, 10..15 gn{1,3,4}{b,g},
// 16..75 tower 't' (sa1,sa2,sa3 x 20 leaves), 76..135 tower 'tg'.
// Per-SA leaf offsets: +0 bnb +1 bng +2 linW +3 linb +4 mlpW +5 mlpb
// +6 mlpbe +7 mlpg, then wn[i]: +8+4i W, +9+4i b, +10+4i be, +11+4i g.
// Biases before BatchNorm (mlpb, wn b, linb) cancel analytically and are
// skipped exactly; GN biases (head) are applied.
// ---------------------------------------------------------------------------
extern "C" void kernel_launch(void* const* d_in, const int* in_sizes, int n_in,
                              void* d_out, int out_size, void* d_ws, size_t ws_size,
                              hipStream_t stream) {
  (void)in_sizes; (void)n_in; (void)out_size; (void)ws_size;
  auto in = [&](int i) { return (const float*)d_in[i]; };

  char* wsp = (char*)d_ws;
  size_t off = 0;
  auto alloc = [&](size_t bytes) -> float* {
    size_t a = (off + 255) & ~(size_t)255;
    off = a + bytes;
    return (float*)(wsp + a);
  };
  float* stbuf = alloc(2 * 256 * sizeof(float));  // persistent BN stats scratch

  auto gemm = [&](const float* A, long long As_m, long long As_k,
                  const float* Bm, long long Bs_k, long long Bs_n,
                  const float* bias, float* D, long long M, int N, int K, int ldD,
                  long long bA, long long bB, long long bD, int nbatch) {
    long long tiles = (M >> 4) * (long long)((N + 15) >> 4);
    dim3 grid((unsigned)CDIV(tiles, 8), (unsigned)nbatch, 1);
    if (As_k == 1 && Bs_k == 1)
      k_gemm_wmma<1, 1><<<grid, 256, 0, stream>>>(A, Bm, bias, D, M, N, K, As_m,
                                                  As_k, Bs_k, Bs_n, ldD, bA, bB, bD);
    else if (As_k == 64 && Bs_k == 16)
      k_gemm_wmma<64, 16><<<grid, 256, 0, stream>>>(A, Bm, bias, D, M, N, K, As_m,
                                                    As_k, Bs_k, Bs_n, ldD, bA, bB, bD);
    else if (As_k == 128 && Bs_k == 16)
      k_gemm_wmma<128, 16><<<grid, 256, 0, stream>>>(A, Bm, bias, D, M, N, K, As_m,
                                                     As_k, Bs_k, Bs_n, ldD, bA, bB, bD);
    else if (As_k == 256 && Bs_k == 16)
      k_gemm_wmma<256, 16><<<grid, 256, 0, stream>>>(A, Bm, bias, D, M, N, K, As_m,
                                                     As_k, Bs_k, Bs_n, ldD, bA, bB, bD);
    else
      k_gemm_wmma<0, 0><<<grid, 256, 0, stream>>>(A, Bm, bias, D, M, N, K, As_m,
                                                  As_k, Bs_k, Bs_n, ldD, bA, bB, bD);
  };
  auto bn_relu = [&](float* X, long long M, int C, const float* g, const float* be) {
    k_zero<<<1, 256, 0, stream>>>(stbuf, 2 * C);
    k_colstats<<<C, 256, 0, stream>>>(X, stbuf, M, C);
    k_bn_apply_relu<<<(unsigned)CDIV(M * C, 256), 256, 0, stream>>>(X, stbuf, g, be, M, C);
  };

  auto run_sa = [&](const float* P, const float* F, int Np, int Cpts, int pb,
                    int npoint, int ns, float bw, bool group_all,
                    float* newxyz, float* fout, int Cout) {
    float* invden = alloc((size_t)B_ * Np * 4);
    k_density<<<CDIV(B_ * Np, 256), 256, 0, stream>>>(P, invden, Np, bw);

    int S, K;
    float *feats, *gxyz, *gden;
    const int Cf = 3 + Cpts;
    if (!group_all) {
      S = npoint; K = ns;
      int* fpsi = (int*)alloc((size_t)B_ * S * 4);
      k_fps<<<B_, Np, 0, stream>>>(P, fpsi, Np, S);
      k_gather3<<<CDIV(B_ * S, 256), 256, 0, stream>>>(P, fpsi, newxyz, Np, S);
      int* knnidx = (int*)alloc((size_t)B_ * S * K * 4);
      k_knn<<<B_ * S, 256, Np * 4, stream>>>(P, newxyz, knnidx, Np, S, K);
      long long Mg = (long long)B_ * S * K;
      feats = alloc((size_t)Mg * Cf * 4);
      gxyz = alloc((size_t)Mg * 3 * 4);
      gden = alloc((size_t)Mg * 4);
      k_group<<<(unsigned)CDIV(Mg, 256), 256, 0, stream>>>(
          P, F, newxyz, invden, knnidx, feats, gxyz, gden, Np, S, K, Cpts, Cf);
    } else {
      S = 1; K = Np;
      long long Mg = (long long)B_ * K;
      feats = alloc((size_t)Mg * Cf * 4);
      gxyz = alloc((size_t)Mg * 3 * 4);
      gden = alloc((size_t)Mg * 4);
      k_group_all<<<(unsigned)CDIV(Mg, 256), 256, 0, stream>>>(
          P, F, invden, feats, gxyz, gden, Np, Cpts, Cf);
    }
    const long long M = (long long)B_ * S * K;

    // MLP: h[M, Cout] = feats[M, Cf] x mlpW[Cout, Cf]^T ; BN+ReLU; density scale
    float* h = alloc((size_t)M * Cout * 4);
    gemm(feats, Cf, 1, in(pb + 4), 1, Cf, nullptr, h, M, Cout, Cf, Cout, 0, 0, 0, 1);
    bn_relu(h, M, Cout, in(pb + 7), in(pb + 6));
    k_den_scale<<<B_ * S, 256, 0, stream>>>(h, gden, K, Cout);

    // WeightNet: gxyz(3) -> 8 -> 8 -> 16 with BN+ReLU each
    float* w1 = alloc((size_t)M * 8 * 4);
    gemm(gxyz, 3, 1, in(pb + 8), 1, 3, nullptr, w1, M, 8, 3, 8, 0, 0, 0, 1);
    bn_relu(w1, M, 8, in(pb + 11), in(pb + 10));
    float* w2 = alloc((size_t)M * 8 * 4);
    gemm(w1, 8, 1, in(pb + 12), 1, 8, nullptr, w2, M, 8, 8, 8, 0, 0, 0, 1);
    bn_relu(w2, M, 8, in(pb + 15), in(pb + 14));
    float* w3 = alloc((size_t)M * 16 * 4);
    gemm(w2, 8, 1, in(pb + 16), 1, 8, nullptr, w3, M, 16, 8, 16, 0, 0, 0, 1);
    bn_relu(w3, M, 16, in(pb + 19), in(pb + 18));

    // einsum bskc,bskw->bscw : per (b,s) batch D[Cout,16] = h^T(Cout,K) x w3(K,16)
    float* fb = alloc((size_t)B_ * S * Cout * 16 * 4);
    gemm(h, /*As_m=*/1, /*As_k=*/Cout, w3, /*Bs_k=*/16, /*Bs_n=*/1, nullptr, fb,
         Cout, 16, K, 16, (long long)K * Cout, (long long)K * 16,
         (long long)Cout * 16, B_ * S);

    // linear: fout[B*S, Cout] = fb[B*S, 16*Cout] x linW[Cout, 16*Cout]^T ; BN+ReLU
    gemm(fb, (long long)16 * Cout, 1, in(pb + 2), 1, (long long)16 * Cout, nullptr,
         fout, (long long)B_ * S, Cout, 16 * Cout, Cout, 0, 0, 0, 1);
    bn_relu(fout, (long long)B_ * S, Cout, in(pb + 1), in(pb + 0));
  };

  auto tower = [&](const float* chw, int Cin, int tb, float* out256) {
    size_t mark = off;
    float* F0 = alloc((size_t)B_ * N_ * Cin * 4);
    k_chw_to_hwc_sel<<<(unsigned)CDIV((long long)B_ * N_ * Cin, 256), 256, 0, stream>>>(
        chw, F0, Cin, Cin, N_);
    float* P0 = alloc((size_t)B_ * N_ * 3 * 4);
    k_chw_to_hwc_sel<<<(unsigned)CDIV((long long)B_ * N_ * 3, 256), 256, 0, stream>>>(
        chw, P0, Cin, 3, N_);
    float* x1 = alloc((size_t)B_ * 512 * 3 * 4);
    float* f1 = alloc((size_t)B_ * 512 * 64 * 4);
    run_sa(P0, F0, N_, Cin, tb + 0, 512, 32, 0.1f, false, x1, f1, 64);
    float* x2 = alloc((size_t)B_ * 128 * 3 * 4);
    float* f2 = alloc((size_t)B_ * 128 * 128 * 4);
    run_sa(x1, f1, 512, 64, tb + 20, 128, 64, 0.2f, false, x2, f2, 128);
    run_sa(x2, f2, 128, 128, tb + 40, 1, 0, 0.4f, true, nullptr, out256, 256);
    off = mark;  // release tower-local scratch
  };

  float* xt = alloc((size_t)B_ * 256 * 4);
  float* gt = alloc((size_t)B_ * 256 * 4);
  tower(in(0), 5, 16, xt);   // tower 't' on xyz (5 channels)
  tower(in(1), 3, 76, gt);   // tower 'tg' on xyz_goal (3 channels)

  // Head: concat -> fc1+GN+ReLU -> fc3+GN+ReLU -> fc4+GN+ReLU -> fc5
  float* xcat = alloc((size_t)B_ * 512 * 4);
  k_concat<<<CDIV(B_ * 512, 256), 256, 0, stream>>>(xt, gt, xcat);
  float* h1 = alloc((size_t)B_ * 256 * 4);
  gemm(xcat, 512, 1, in(2), 1, 512, in(3), h1, B_, 256, 512, 256, 0, 0, 0, 1);
  k_gn_relu<<<B_, 256, 0, stream>>>(h1, in(11), in(10), 256);
  float* h3 = alloc((size_t)B_ * 128 * 4);
  gemm(h1, 256, 1, in(4), 1, 256, in(5), h3, B_, 128, 256, 128, 0, 0, 0, 1);
  k_gn_relu<<<B_, 256, 0, stream>>>(h3, in(13), in(12), 128);
  float* h4 = alloc((size_t)B_ * 64 * 4);
  gemm(h3, 128, 1, in(6), 1, 128, in(7), h4, B_, 64, 128, 64, 0, 0, 0, 1);
  k_gn_relu<<<B_, 256, 0, stream>>>(h4, in(15), in(14), 64);
  k_fc5<<<CDIV(B_ * 6, 256), 256, 0, stream>>>(h4, in(8), in(9), (float*)d_out);
}